// SemanticAttentionBatched_9225589752129
// MI455X (gfx1250) — compile-verified
//
#include <hip/hip_runtime.h>

// Semantic attention, fused single kernel for gfx1250 (MI455X).
// h:[R,N,D] f32, w1:[R,D,A] f32, w2:[R,A,1] f32 -> out:[N,D] f32
// R=8, D=128, A=64.  Block = 256 threads = 8 waves; wave w == relation r.
// Scoring GEMM via V_WMMA_F32_16X16X4_F32; h tiles staged to LDS via the
// Tensor Data Mover (tensor_load_to_lds) with hardware row padding.

#define RR   8
#define DD   128
#define AA   64
#define MT   64      // nodes per block
#define ROWP 132     // LDS row stride in floats (128 data + 4 pad, TDM pad)

typedef float v2f __attribute__((ext_vector_type(2)));
typedef float v8f __attribute__((ext_vector_type(8)));
typedef unsigned int v4u __attribute__((ext_vector_type(4)));
typedef int          v4i __attribute__((ext_vector_type(4)));
typedef int          v8i __attribute__((ext_vector_type(8)));

#if __has_builtin(__builtin_amdgcn_tensor_load_to_lds) && \
    __has_builtin(__builtin_amdgcn_s_wait_tensorcnt)
#define HAVE_TDM 1
#if __clang_major__ >= 23
#define TDM_LOAD(g0, g1, g2, g3) \
    __builtin_amdgcn_tensor_load_to_lds((g0), (g1), (g2), (g3), \
        (v8i){0, 0, 0, 0, 0, 0, 0, 0}, 0)
#else
#define TDM_LOAD(g0, g1, g2, g3) \
    __builtin_amdgcn_tensor_load_to_lds((g0), (g1), (g2), (g3), 0)
#endif
#endif

__device__ __forceinline__ float sigf(float x) {
    return 1.0f / (1.0f + __expf(-x));
}

template <bool USE_W1T>
__global__ __launch_bounds__(256)
void semattn_fused_kernel(const float* __restrict__ h,
                          const float* __restrict__ w1,
                          const float* __restrict__ w2,
                          const float* __restrict__ w1t,  // [R][A][D] transposed
                          float* __restrict__ out,
                          int N)
{
    extern __shared__ float smem[];
    float* hT = smem;                       // [RR][MT][ROWP] staged h tiles
    float* sc = hT + RR * MT * ROWP;        // [RR][MT] raw scores
    float* at = sc + RR * MT;               // [RR][MT] softmax weights

    const int tid    = threadIdx.x;
    const int wave   = tid >> 5;            // relation id for this wave
    const int lane   = tid & 31;
    const int lane15 = lane & 15;
    const int half   = lane >> 4;           // 0: lanes 0-15, 1: lanes 16-31
    const int r      = wave;
    const int n0     = blockIdx.x * MT;

    // ---------------- Stage h[r, n0:n0+64, :] into LDS ----------------------
#ifdef HAVE_TDM
    {
        // Per-wave Tensor Data Mover load: 64 rows x 512B, contiguous in
        // global, padded +4 dwords per 128 dwords in LDS (-> ROWP=132).
        // OOB rows (tail tile) read as zero; stores are guarded later.
        const int rs = __builtin_amdgcn_readfirstlane(r);
        unsigned lds_off = (unsigned)(unsigned long long)(unsigned long long)
                           (uintptr_t)(void*)(hT + rs * MT * ROWP);
        lds_off = (unsigned)__builtin_amdgcn_readfirstlane((int)lds_off);
        unsigned long long ga =
            (unsigned long long)(uintptr_t)(h + ((size_t)rs * N + n0) * DD);
        unsigned ga_lo = (unsigned)__builtin_amdgcn_readfirstlane((int)(ga & 0xffffffffu));
        unsigned ga_hi = (unsigned)__builtin_amdgcn_readfirstlane((int)((ga >> 32) & 0x1ffffffu));
        unsigned td1   = (unsigned)__builtin_amdgcn_readfirstlane(N - n0); // rows avail

        v4u g0;
        g0[0] = 1u;                           // count=1, user descriptor
        g0[1] = lds_off;                      // lds_addr (bytes)
        g0[2] = ga_lo;                        // global_addr[31:0]
        g0[3] = ga_hi | (2u << 30);           // global_addr[56:32] | type=2

        v8i g1;
        g1[0] = (int)((2u << 16)              // data_size = 4B
                    | (1u << 20)              // pad_enable
                    | (6u << 22)              // pad_interval: 128 dwords
                    | (3u << 25));            // pad_amount:   4 dwords
        g1[1] = (int)(128u << 16);            // tensor_dim0[15:0]=128 (bits 63:48)
        g1[2] = (int)((td1 & 0xffffu) << 16); // tensor_dim1[15:0]   (bits 95:80)
        g1[3] = (int)(((td1 >> 16) & 0xffffu) // tensor_dim1[31:16]
                    | (128u << 16));          // tile_dim0=128       (bits 127:112)
        g1[4] = (int)64u;                     // tile_dim1=64; tile_dim2=0
        g1[5] = (int)128u;                    // tensor_dim0_stride[31:0]=128
        g1[6] = 0;                            // dim0_stride[47:32], dim1_stride lo
        g1[7] = 0;                            // dim1_stride hi (unused for 2D)

        v4i gz = {0, 0, 0, 0};
        TDM_LOAD(g0, g1, gz, gz);
        __builtin_amdgcn_s_wait_tensorcnt(0);
    }
#else
    {
        const int d4 = lane * 4;              // 32 lanes x float4 = 128 floats
        for (int row = 0; row < MT; ++row) {
            int n = n0 + row; if (n > N - 1) n = N - 1;     // clamp tail
            const float4 v = *(const float4*)(h + ((size_t)r * N + n) * DD + d4);
            *(float4*)(hT + (r * MT + row) * ROWP + d4) = v;
        }
    }
#endif
    __syncthreads();

    // ---------------- Scoring GEMM: hidden = Htile[64x128] * W1_r[128x64] ---
    v8f c[4][4];
#pragma unroll
    for (int mt = 0; mt < 4; ++mt)
#pragma unroll
        for (int nt = 0; nt < 4; ++nt)
            c[mt][nt] = (v8f){0.f, 0.f, 0.f, 0.f, 0.f, 0.f, 0.f, 0.f};

    const float* w1r  = w1 + (size_t)r * DD * AA;    // [D][A] row-major
    const float* w1tr = USE_W1T ? (w1t + (size_t)r * AA * DD) : nullptr; // [A][D]

    for (int k0 = 0; k0 < DD; k0 += 4) {
        const int kk = k0 + half * 2;        // K split across lane halves
        // A fragments (16x4 f32) from LDS
        v2f a[4];
#pragma unroll
        for (int mt = 0; mt < 4; ++mt) {
            const float* p = hT + (r * MT + mt * 16 + lane15) * ROWP + kk;
            a[mt] = *(const v2f*)p;          // ds_load_b64
        }
        // B fragments (4x16 f32) from W1 (L2-resident)
        v2f b[4];
#pragma unroll
        for (int nt = 0; nt < 4; ++nt) {
            const int col = nt * 16 + lane15;
            if (USE_W1T) {
                b[nt] = *(const v2f*)(w1tr + (size_t)col * DD + kk);  // b64
            } else {
                v2f t;
                t.x = w1r[(size_t)kk * AA + col];
                t.y = w1r[(size_t)(kk + 1) * AA + col];
                b[nt] = t;
            }
        }
#pragma unroll
        for (int mt = 0; mt < 4; ++mt)
#pragma unroll
            for (int nt = 0; nt < 4; ++nt)
                c[mt][nt] = __builtin_amdgcn_wmma_f32_16x16x4_f32(
                    false, a[mt], false, b[nt], (short)0, c[mt][nt],
                    false, false);
    }

    // ------------- hidden -> sigmoid -> dot(w2) -> per-row score -----------
    float w2v[4];
    const float* w2r = w2 + (size_t)r * AA;
#pragma unroll
    for (int nt = 0; nt < 4; ++nt) w2v[nt] = w2r[nt * 16 + lane15];

#pragma unroll
    for (int mt = 0; mt < 4; ++mt) {
#pragma unroll
        for (int i = 0; i < 8; ++i) {
            float acc = 0.f;
#pragma unroll
            for (int nt = 0; nt < 4; ++nt)
                acc += sigf(c[mt][nt][i]) * w2v[nt];
            acc += __shfl_xor(acc, 1);
            acc += __shfl_xor(acc, 2);
            acc += __shfl_xor(acc, 4);
            acc += __shfl_xor(acc, 8);
            if (lane15 == 0)
                sc[r * MT + mt * 16 + i + half * 8] = acc;
        }
    }
    __syncthreads();

    // ---------------- Softmax over relations, per node ---------------------
    if (tid < MT) {
        float s[RR];
        float mx = -3.402823466e38f;
#pragma unroll
        for (int rr = 0; rr < RR; ++rr) {
            s[rr] = sc[rr * MT + tid];
            mx = fmaxf(mx, s[rr]);
        }
        float sum = 0.f;
#pragma unroll
        for (int rr = 0; rr < RR; ++rr) {
            s[rr] = __expf(s[rr] - mx);
            sum += s[rr];
        }
        const float inv = 1.0f / sum;
#pragma unroll
        for (int rr = 0; rr < RR; ++rr)
            at[rr * MT + tid] = s[rr] * inv;
    }
    __syncthreads();

    // ---------------- Weighted sum over relations, from LDS ----------------
    {
        const int d4 = lane * 4;
#pragma unroll
        for (int j = 0; j < 8; ++j) {
            const int m = wave * 8 + j;
            const int n = n0 + m;
            if (n < N) {
                float4 accv = {0.f, 0.f, 0.f, 0.f};
#pragma unroll
                for (int rr = 0; rr < RR; ++rr) {
                    const float aw = at[rr * MT + m];
                    const float4 hv = *(const float4*)(hT + (rr * MT + m) * ROWP + d4);
                    accv.x += aw * hv.x;
                    accv.y += aw * hv.y;
                    accv.z += aw * hv.z;
                    accv.w += aw * hv.w;
                }
                *(float4*)(out + (size_t)n * DD + d4) = accv;
            }
        }
    }
}

// One-shot transpose of w1 [R][D][A] -> w1t [R][A][D] (256 KB, L2-resident).
__global__ void w1_transpose_kernel(const float* __restrict__ w1,
                                    float* __restrict__ w1t) {
    const int idx = blockIdx.x * blockDim.x + threadIdx.x;   // over R*D*A
    if (idx < RR * DD * AA) {
        const int a = idx % AA;
        const int k = (idx / AA) % DD;
        const int r = idx / (AA * DD);
        w1t[((size_t)r * AA + a) * DD + k] = w1[idx];
    }
}

extern "C" void kernel_launch(void* const* d_in, const int* in_sizes, int n_in,
                              void* d_out, int out_size, void* d_ws, size_t ws_size,
                              hipStream_t stream) {
    const float* h  = (const float*)d_in[0];   // [R, N, D]
    const float* w1 = (const float*)d_in[1];   // [R, D, A]
    const float* w2 = (const float*)d_in[2];   // [R, A, 1]
    float* out = (float*)d_out;                // [N, D]

    const int N = in_sizes[0] / (RR * DD);
    const int blocks = (N + MT - 1) / MT;
    const size_t shmem = (size_t)(RR * MT * ROWP + 2 * RR * MT) * sizeof(float);
    const size_t w1t_bytes = (size_t)RR * DD * AA * sizeof(float);

    if (ws_size >= w1t_bytes) {
        float* w1t = (float*)d_ws;
        w1_transpose_kernel<<<(RR * DD * AA + 255) / 256, 256, 0, stream>>>(w1, w1t);
        semattn_fused_kernel<true><<<dim3(blocks), dim3(256), shmem, stream>>>(
            h, w1, w2, w1t, out, N);
    } else {
        semattn_fused_kernel<false><<<dim3(blocks), dim3(256), shmem, stream>>>(
            h, w1, w2, nullptr, out, N);
    }
}